// LDS_67259187855624
// MI455X (gfx1250) — compile-verified
//
#include <hip/hip_runtime.h>

// LDS sampleX on MI455X (gfx1250): stride-16 recurrence + log-doubling causal
// conv, all matrix work on V_WMMA_F32_16X16X4_F32 (fp32 tensor path).
//
// x[t] = A x[t-1] + b[t]  ==>  x[t] = A^16 x[t-16] + c[t],
//   c[t] = sum_{j<16} A^j b[t-j]   (4 doubling GEMM levels, fully parallel)
// 6250 chunks x 160 rows; zero-carry local scan per chunk; carry across chunks
// is exact to fp32 with 2 terms because ||A^320|| underflows to 0.

#define XDIM   64
#define NSTEP  1000000
#define ROWS   160
#define HALO   16
#define RTOT   (ROWS + HALO)        // 176
#define BPC    (ROWS / 16)          // 10 blocks per chunk
#define NCHUNK (NSTEP / ROWS)       // 6250 (exact)
#define LDW    68                   // padded LDS row stride (floats)

// workspace layout (floats)
#define WS_P1  0
#define WS_P2  4096
#define WS_P4  8192
#define WS_P8  12288
#define WS_P16 16384
#define WS_J   20480                      // J_m = A^(16m), m=1..10
#define WS_F   (WS_J + 10 * 4096)         // F[c]: 16x64 local final rows

typedef float v2f __attribute__((ext_vector_type(2)));
typedef float v8f __attribute__((ext_vector_type(8)));

#if defined(__HIP_DEVICE_COMPILE__)
#if !__has_builtin(__builtin_amdgcn_wmma_f32_16x16x4_f32)
#error "missing __builtin_amdgcn_wmma_f32_16x16x4_f32 on this toolchain"
#endif
#endif

__device__ __forceinline__ v8f wmma4(v2f a, v2f b, v8f c) {
  // 8 args: (neg_a, A, neg_b, B, c_mod, C, reuse_a, reuse_b)
  return __builtin_amdgcn_wmma_f32_16x16x4_f32(false, a, false, b, (short)0, c,
                                               false, false);
}

// A-operand 16x4 f32: lanes 0-15 -> (K0,K1) of row M=lane; lanes 16-31 -> (K2,K3)
__device__ __forceinline__ v2f ldA(const float* X, int lane, int k0) {
  const float* p = X + (lane & 15) * LDW + k0 + (lane >> 4) * 2;
  v2f a = {p[0], p[1]};
  return a;
}
// B-operand 4x16 for Y = X @ Mat^T: B[k][n] = Mat[n][k]; Mat staged row-major.
__device__ __forceinline__ v2f ldB(const float* M, int lane, int k0, int n0) {
  const float* p = M + (n0 + (lane & 15)) * LDW + k0 + (lane >> 4) * 2;
  v2f b = {p[0], p[1]};
  return b;
}
// C/D 16x16 f32: VGPR v holds row v (lanes 0-15) / row v+8 (lanes 16-31).
__device__ __forceinline__ v8f ldC(const float* src, int lane, int n0) {
  const int col = n0 + (lane & 15);
  const int r0 = (lane >> 4) * 8;
  v8f c;
#pragma unroll
  for (int v = 0; v < 8; ++v) c[v] = src[(r0 + v) * LDW + col];
  return c;
}
__device__ __forceinline__ void stC(float* dst, int lane, int n0, v8f c) {
  const int col = n0 + (lane & 15);
  const int r0 = (lane >> 4) * 8;
#pragma unroll
  for (int v = 0; v < 8; ++v) dst[(r0 + v) * LDW + col] = c[v];
}

// Cache the wave's 16 B fragments in VGPRs (reused across all row-blocks).
__device__ __forceinline__ void load_breg(v2f* breg, const float* M, int lane,
                                          int n0) {
#pragma unroll
  for (int kk = 0; kk < 16; ++kk) breg[kk] = ldB(M, lane, kk * 4, n0);
}
__device__ __forceinline__ v8f gemm_breg(const float* X, const v2f* breg,
                                         int lane, v8f acc) {
#pragma unroll
  for (int kk = 0; kk < 16; ++kk) acc = wmma4(ldA(X, lane, kk * 4), breg[kk], acc);
  return acc;
}

// copy a row-major 64x64 matrix from global into LDS (stride LDW), float4s
__device__ __forceinline__ void stage_mat(float* M, const float* g, int tid,
                                          int nthr) {
  for (int i = tid; i < 64 * 16; i += nthr) {
    int r = i >> 4, q = i & 15;
    *(float4*)(M + r * LDW + q * 4) = ((const float4*)(g + r * 64))[q];
  }
}

__device__ __forceinline__ void gsync() {
  __threadfence();
  __syncthreads();
}
__device__ __forceinline__ void mm64(float* D, const float* X, const float* Y) {
  for (int e = threadIdx.x; e < 4096; e += blockDim.x) {
    int r = e >> 6, c = e & 63;
    float s = 0.f;
    for (int k = 0; k < 64; ++k) s += X[r * 64 + k] * Y[k * 64 + c];
    D[e] = s;
  }
}

// ---- Kernel 1: tiny precompute of matrix powers (1 block) ----
__global__ __launch_bounds__(256) void lds_precompute(const float* __restrict__ A,
                                                      float* __restrict__ ws) {
  for (int i = threadIdx.x; i < 4096; i += blockDim.x) ws[WS_P1 + i] = A[i];
  gsync();
  mm64(ws + WS_P2, ws + WS_P1, ws + WS_P1);
  gsync();
  mm64(ws + WS_P4, ws + WS_P2, ws + WS_P2);
  gsync();
  mm64(ws + WS_P8, ws + WS_P4, ws + WS_P4);
  gsync();
  mm64(ws + WS_P16, ws + WS_P8, ws + WS_P8);
  gsync();
  for (int i = threadIdx.x; i < 4096; i += blockDim.x)
    ws[WS_J + i] = ws[WS_P16 + i];                 // J_1 = A^16
  gsync();
  for (int m = 1; m < BPC; ++m) {                  // J_{m+1} = J_m * A^16
    mm64(ws + WS_J + m * 4096, ws + WS_J + (m - 1) * 4096, ws + WS_P16);
    gsync();
  }
}

// ---- Kernel 2: per-chunk local scan (grid-parallel) ----
__global__ __launch_bounds__(128) void lds_chunk_local(
    const float* __restrict__ norm, const float* __restrict__ QChol,
    const float* __restrict__ Q0Chol, const float* __restrict__ x0,
    float* __restrict__ out, float* __restrict__ ws) {
  extern __shared__ float smem[];
  float* buf0 = smem;                       // 176 x 68
  float* buf1 = smem + RTOT * LDW;          // 176 x 68
  float* Mslot = smem + 2 * RTOT * LDW;     // 64 x 68

  const int tid = threadIdx.x;
  const int lane = tid & 31;                // wave32
  const int n0 = (tid >> 5) * 16;           // this wave's 16-col tile
  const int c = blockIdx.x;

  // Phase 1: stage eps rows (16-row halo) + QChol
  for (int i = tid; i < RTOT * 16; i += blockDim.x) {
    int r = i >> 4, q = i & 15;
    long t = (long)c * ROWS - HALO + r;
    float4 v = make_float4(0.f, 0.f, 0.f, 0.f);
    if (t >= 0) v = ((const float4*)(norm + (size_t)t * XDIM))[q];
    *(float4*)(buf0 + r * LDW + q * 4) = v;
  }
  stage_mat(Mslot, QChol, tid, blockDim.x);
  __syncthreads();

  // Phase 2: drive b = eps @ QChol^T
  {
    v2f breg[16];
    load_breg(breg, Mslot, lane, n0);
    for (int rb = 0; rb < RTOT / 16; ++rb) {
      v8f acc = {};
      acc = gemm_breg(buf0 + rb * 16 * LDW, breg, lane, acc);
      stC(buf1 + rb * 16 * LDW, lane, n0, acc);
    }
  }
  __syncthreads();

  // Phase 3: b[0] = x0 + eps0 @ Q0Chol^T (t==0 lives only in chunk 0, r=HALO)
  if (c == 0 && tid < XDIM) {
    float s = x0[tid];
    for (int k = 0; k < XDIM; ++k) s += norm[k] * Q0Chol[tid * XDIM + k];
    buf1[HALO * LDW + tid] = s;
  }
  __syncthreads();

  // Phase 4: 4 doubling levels -> c[t] = sum_{j<16} A^j b[t-j]
  float* src = buf1;
  float* dst = buf0;
  const int pw[4] = {WS_P1, WS_P2, WS_P4, WS_P8};
#pragma unroll
  for (int lvl = 0; lvl < 4; ++lvl) {
    stage_mat(Mslot, ws + pw[lvl], tid, blockDim.x);
    __syncthreads();
    v2f breg[16];
    load_breg(breg, Mslot, lane, n0);
    const int d = 1 << lvl;
    for (int rb = 0; rb < RTOT / 16; ++rb) {
      v8f acc = ldC(src + rb * 16 * LDW, lane, n0);
      const int srow = rb * 16 + (lane & 15) - d;
      const bool ok = srow >= 0;
      const float* Xr = src + srow * LDW + (lane >> 4) * 2;
#pragma unroll
      for (int kk = 0; kk < 16; ++kk) {
        v2f a = {0.f, 0.f};
        if (ok) { a.x = Xr[kk * 4]; a.y = Xr[kk * 4 + 1]; }
        acc = wmma4(a, breg[kk], acc);
      }
      stC(dst + rb * 16 * LDW, lane, n0, acc);
    }
    __syncthreads();
    float* t2 = src; src = dst; dst = t2;
  }
  // c[] now in src(==buf1); dst(==buf0) free for x.

  // Phase 5: X_m = X_{m-1} @ (A^16)^T + C_m  (zero carry; 10 sequential blocks)
  stage_mat(Mslot, ws + WS_P16, tid, blockDim.x);
  __syncthreads();
  {
    v2f breg[16];
    load_breg(breg, Mslot, lane, n0);
    for (int m = 0; m < BPC; ++m) {
      const int rbase = HALO + m * 16;
      v8f acc = ldC(src + rbase * LDW, lane, n0);
      if (m > 0) acc = gemm_breg(dst + (rbase - 16) * LDW, breg, lane, acc);
      stC(dst + rbase * LDW, lane, n0, acc);
      __syncthreads();
    }
  }

  // Phase 6: write local x rows; save final 16 rows F_c for carry pass
  for (int i = tid; i < ROWS * 16; i += blockDim.x) {
    int r = i >> 4, q = i & 15;
    size_t t = (size_t)c * ROWS + r;
    *(float4*)(out + t * XDIM + q * 4) =
        *(const float4*)(buf0 + (HALO + r) * LDW + q * 4);
  }
  for (int i = tid; i < 16 * 16; i += blockDim.x) {
    int r = i >> 4, q = i & 15;
    *(float4*)(ws + WS_F + (size_t)c * 1024 + r * 64 + q * 4) =
        *(const float4*)(buf0 + (HALO + ROWS - 16 + r) * LDW + q * 4);
  }
}

// ---- Kernel 3: fp32-exact carry + per-block correction (grid-parallel) ----
__global__ __launch_bounds__(128) void lds_apply_carry(
    float* __restrict__ out, const float* __restrict__ ws) {
  __shared__ float carry[16 * LDW];
  __shared__ float xst[16 * LDW];
  __shared__ float Mslot[64 * LDW];
  const int tid = threadIdx.x;
  const int lane = tid & 31;
  const int n0 = (tid >> 5) * 16;
  const int c = blockIdx.x;
  if (c == 0) return;  // carry into chunk 0 is exactly zero

  // Carry_c = F_{c-1} + F_{c-2} @ (A^160)^T   (A^320 underflows to 0 in fp32)
  for (int i = tid; i < 16 * 16; i += blockDim.x) {
    int r = i >> 4, q = i & 15;
    *(float4*)(carry + r * LDW + q * 4) =
        ((const float4*)(ws + WS_F + (size_t)(c - 1) * 1024 + r * 64))[q];
  }
  __syncthreads();
  if (c >= 2) {
    for (int i = tid; i < 16 * 16; i += blockDim.x) {
      int r = i >> 4, q = i & 15;
      *(float4*)(xst + r * LDW + q * 4) =
          ((const float4*)(ws + WS_F + (size_t)(c - 2) * 1024 + r * 64))[q];
    }
    stage_mat(Mslot, ws + WS_J + 9 * 4096, tid, blockDim.x);  // J_10 = A^160
    __syncthreads();
    v8f acc = ldC(carry, lane, n0);
    {
      v2f breg[16];
      load_breg(breg, Mslot, lane, n0);
      acc = gemm_breg(xst, breg, lane, acc);
    }
    __syncthreads();            // all waves done reading carry
    stC(carry, lane, n0, acc);
    __syncthreads();
  }

  // out rows of block m += Carry_c @ (A^{16(m+1)})^T
  const int col = n0 + (lane & 15);
  const int r0 = (lane >> 4) * 8;
  for (int m = 0; m < BPC; ++m) {
    __syncthreads();            // previous Mslot readers done
    stage_mat(Mslot, ws + WS_J + m * 4096, tid, blockDim.x);
    __syncthreads();
    v2f breg[16];
    load_breg(breg, Mslot, lane, n0);
    const size_t t0 = (size_t)c * ROWS + (size_t)m * 16;
    v8f acc;
#pragma unroll
    for (int v = 0; v < 8; ++v) acc[v] = out[(t0 + r0 + v) * XDIM + col];
    acc = gemm_breg(carry, breg, lane, acc);
#pragma unroll
    for (int v = 0; v < 8; ++v) out[(t0 + r0 + v) * XDIM + col] = acc[v];
  }
}

extern "C" void kernel_launch(void* const* d_in, const int* in_sizes, int n_in,
                              void* d_out, int out_size, void* d_ws,
                              size_t ws_size, hipStream_t stream) {
  (void)in_sizes; (void)n_in; (void)out_size; (void)ws_size;
  const float* norm   = (const float*)d_in[0];
  const float* A      = (const float*)d_in[1];
  const float* QChol  = (const float*)d_in[2];
  const float* Q0Chol = (const float*)d_in[3];
  const float* x0     = (const float*)d_in[4];
  float* out = (float*)d_out;
  float* ws  = (float*)d_ws;

  lds_precompute<<<1, 256, 0, stream>>>(A, ws);

  const size_t shbytes = (size_t)(2 * RTOT * LDW + 64 * LDW) * sizeof(float);
  (void)hipFuncSetAttribute(reinterpret_cast<const void*>(lds_chunk_local),
                            hipFuncAttributeMaxDynamicSharedMemorySize,
                            (int)shbytes);
  lds_chunk_local<<<NCHUNK, 128, shbytes, stream>>>(norm, QChol, Q0Chol, x0,
                                                    out, ws);
  lds_apply_carry<<<NCHUNK, 128, 0, stream>>>(out, ws);
}